// NeuralRenderer_11201274708292
// MI455X (gfx1250) — compile-verified
//
#include <hip/hip_runtime.h>

typedef float v2f __attribute__((ext_vector_type(2)));
typedef float v8f __attribute__((ext_vector_type(8)));
typedef int   v4i __attribute__((ext_vector_type(4)));

#define AS1 __attribute__((address_space(1)))
#define AS3 __attribute__((address_space(3)))

#define IMG 256
#define CHB 512          // faces per LDS buffer (mult of 16); 512*12 floats = 24576 B
                         // per chunk: 512*3 = 1536 float4 = 6 b128 loads per thread (256 thr)

#if __has_builtin(__builtin_amdgcn_global_load_async_to_lds_b128) && \
    __has_builtin(__builtin_amdgcn_s_wait_asynccnt)
#define USE_ASYNC 1
#else
#define USE_ASYNC 0
#endif

// ---------------------------------------------------------------------------
// Kernel 1: per-face preprocessing.
// Projects the 3 vertices, computes lighting, and emits a 12-float record in
// "WMMA-B-ready" layout (6 float2 slots):
//   [ (a0,b0) (a1,b1) (az,bz) (c0,0) (c1,0) (cz,0) ]
// so lane L<16 (rows K0,K1) loads slot e, lane L>=16 (rows K2,K3) loads slot 3+e.
// Degenerate / padded faces get a sentinel (w0 == -1 everywhere -> never inside).
// ---------------------------------------------------------------------------
__global__ __launch_bounds__(256) void nr_prep(
    const float* __restrict__ verts, const int* __restrict__ faces,
    const float* __restrict__ cams, float* __restrict__ rec,
    float* __restrict__ lightOut, int B, int V, int F, int Fpad)
{
  int i = blockIdx.x * blockDim.x + threadIdx.x;
  if (i >= B * Fpad) return;
  int b = i / Fpad, f = i - b * Fpad;
  float* r = rec + (size_t)i * 12;

  if (f >= F) {
    r[0]=0.f; r[1]=0.f; r[2]=0.f; r[3]=0.f; r[4]=0.f; r[5]=0.f;
    r[6]=-1.f; r[7]=0.f; r[8]=0.f; r[9]=0.f; r[10]=0.f; r[11]=0.f;
    lightOut[i] = 0.f;
    return;
  }

  const float* cam = cams + b * 8;
  float fs = cam[0];
  float tx = cam[1], ty = cam[2], tz = cam[3];
  float qw = cam[4], qx = cam[5], qy = cam[6], qz = cam[7];

  float X[3][3];
  #pragma unroll
  for (int k = 0; k < 3; ++k) {
    int vi = faces[f * 3 + k];
    const float* vp = verts + ((size_t)b * V + vi) * 3;
    float x = vp[0], y = vp[1], z = vp[2];
    // quat rotate: X + 2*(w*cross(v,X) + cross(v,cross(v,X)))
    float uvx = qy*z - qz*y, uvy = qz*x - qx*z, uvz = qx*y - qy*x;
    float uux = qy*uvz - qz*uvy, uuy = qz*uvx - qx*uvz, uuz = qx*uvy - qy*uvx;
    float rx = x + 2.f*(qw*uvx + uux);
    float ry = y + 2.f*(qw*uvy + uuy);
    float rz = z + 2.f*(qw*uvz + uuz);
    float Xtx = rx + tx, Xty = ry + ty, Xtz = rz + tz;
    // z_proj = Xtz + OFFSET_Z; verts_cam adds +1 back -> z_cam = Xtz
    X[k][0] = fs * Xtx / Xtz;
    X[k][1] = fs * Xty / Xtz;
    X[k][2] = Xtz;
  }

  // lighting: n = normalize(cross(e1,e2)); light = 0.8 + 0.5*max(n.y, 0)
  float e1x = X[1][0]-X[0][0], e1y = X[1][1]-X[0][1], e1z = X[1][2]-X[0][2];
  float e2x = X[2][0]-X[0][0], e2y = X[2][1]-X[0][1], e2z = X[2][2]-X[0][2];
  float nx = e1y*e2z - e1z*e2y;
  float ny = e1z*e2x - e1x*e2z;
  float nz = e1x*e2y - e1y*e2x;
  float nn = sqrtf(nx*nx + ny*ny + nz*nz) + 1e-8f;
  lightOut[i] = 0.8f + 0.5f * fmaxf(ny / nn, 0.f);

  float x0 = X[0][0], y0 = X[0][1], x1 = X[1][0], y1 = X[1][1], x2 = X[2][0], y2 = X[2][1];
  float z0 = X[0][2], z1 = X[1][2], z2 = X[2][2];
  float den = (y1 - y2) * (x0 - x2) + (x2 - x1) * (y0 - y2);
  if (fabsf(den) > 1e-9f) {
    float inv = 1.f / den;
    float a0 = (y1 - y2) * inv, b0 = (x2 - x1) * inv;
    float c0 = -(a0 * x2 + b0 * y2);
    float a1 = (y2 - y0) * inv, b1 = (x0 - x2) * inv;
    float c1 = -(a1 * x2 + b1 * y2);
    float dz0 = z0 - z2, dz1 = z1 - z2;
    float az = a0 * dz0 + a1 * dz1;
    float bz = b0 * dz0 + b1 * dz1;
    float cz = z2 + c0 * dz0 + c1 * dz1;
    r[0]=a0; r[1]=b0; r[2]=a1; r[3]=b1; r[4]=az; r[5]=bz;
    r[6]=c0; r[7]=0.f; r[8]=c1; r[9]=0.f; r[10]=cz; r[11]=0.f;
  } else {
    r[0]=0.f; r[1]=0.f; r[2]=0.f; r[3]=0.f; r[4]=0.f; r[5]=0.f;
    r[6]=-1.f; r[7]=0.f; r[8]=0.f; r[9]=0.f; r[10]=0.f; r[11]=0.f;
  }
}

// ---------------------------------------------------------------------------
// Chunk stage-in: 1536 float4s, 6 per thread.
// Async path: global_load_async_to_lds_b128 (ASYNCcnt), bypasses VGPRs.
// Fallback: plain global->VGPR->LDS copy with a prefetch hint.
// ---------------------------------------------------------------------------
__device__ __forceinline__ void issue_chunk(const float4* __restrict__ src,
                                            float4* dst_lds, int tid)
{
  #pragma unroll
  for (int k = 0; k < 6; ++k) {
    int t = tid + k * 256;
#if USE_ASYNC
    __builtin_amdgcn_global_load_async_to_lds_b128(
        (AS1 v4i*)(unsigned long long)(src + t),
        (AS3 v4i*)(unsigned int)(unsigned long long)(dst_lds + t),
        0, 0);
#else
    dst_lds[t] = src[t];
#endif
  }
#if !USE_ASYNC
  __builtin_prefetch((const void*)(src + CHB * 3 + tid * 4), 0, 3);
#endif
}

// ---------------------------------------------------------------------------
// Kernel 2: WMMA rasterizer. One wave32 owns 16 pixels (one image row segment).
// Per 16-face group: 3x V_WMMA_F32_16X16X4_F32 produce w0, w1, z for the
// 16x16 pixel-face tile (edge functions are affine in pixel coords -> rank-4
// GEMM on the f32 matrix pipe). Face records stream through LDS with
// double-buffered async copies; per-pixel argmin finishes with a shfl_xor
// tree inside each 16-lane half (tie-break: lowest face id).
// ---------------------------------------------------------------------------
__global__ __launch_bounds__(256) void nr_raster(
    const float* __restrict__ rec, int* __restrict__ fidOut, int Fpad)
{
  __shared__ float4 ldsBuf[2][CHB * 3];          // 2 x 24576 B
  const int P = IMG * IMG;
  const int tid  = threadIdx.x;
  const int lane = tid & 31;
  const int wave = tid >> 5;
  const int col  = lane & 15;
  const bool hi  = lane >= 16;

  const int blocksPerB = P / (16 * 8);           // 512
  const int b       = blockIdx.x / blocksPerB;
  const int pgInB   = (blockIdx.x % blocksPerB) * 8 + wave;
  const int basePix = pgInB * 16;                // 16 consecutive pixels, one row

  // A matrix (16x4): rows = pixels (px, py, 1, 0). Lanes 0-15 hold K0/K1,
  // lanes 16-31 hold K2/K3 (constant 1, 0).
  v2f A;
  if (!hi) {
    int p = basePix + lane;
    A.x = (2.0f * (float)(p & (IMG - 1)) + 1.0f) * (1.0f / IMG) - 1.0f;
    A.y = 1.0f - (2.0f * (float)(p >> 8) + 1.0f) * (1.0f / IMG);
  } else {
    A.x = 1.0f; A.y = 0.0f;
  }

  float zmin[8]; int fid[8];
  #pragma unroll
  for (int q = 0; q < 8; ++q) { zmin[q] = __builtin_inff(); fid[q] = -1; }

  const float4* srcBase = (const float4*)rec + (size_t)b * Fpad * 3;
  const int nCh = Fpad / CHB;                    // all chunks full (Fpad % CHB == 0)

  issue_chunk(srcBase, ldsBuf[0], tid);          // preload chunk 0

  for (int c = 0; c < nCh; ++c) {
    const int cur = c & 1;
    if (c + 1 < nCh) {
      issue_chunk(srcBase + (size_t)(c + 1) * CHB * 3, ldsBuf[1 - cur], tid);
#if USE_ASYNC
      __builtin_amdgcn_s_wait_asynccnt(6);       // my chunk-c loads landed (in-order),
                                                 // chunk-(c+1)'s 6 may still fly
#endif
    } else {
#if USE_ASYNC
      __builtin_amdgcn_s_wait_asynccnt(0);
#endif
    }
    __syncthreads();                             // all waves' chunk-c data visible

    const float2* l2 = (const float2*)ldsBuf[cur];
    #pragma unroll 2
    for (int g = 0; g < CHB / 16; ++g) {
      const int fl = g * 16 + col;
      const int e  = fl * 6 + (hi ? 3 : 0);
      float2 cB0 = l2[e + 0];   // edge0: (a0,b0) / (c0,0)
      float2 cB1 = l2[e + 1];   // edge1: (a1,b1) / (c1,0)
      float2 cB2 = l2[e + 2];   // zplane:(az,bz) / (cz,0)
      v2f B0; B0.x = cB0.x; B0.y = cB0.y;
      v2f B1; B1.x = cB1.x; B1.y = cB1.y;
      v2f B2; B2.x = cB2.x; B2.y = cB2.y;

      v8f z8 = {};
      v8f W0 = __builtin_amdgcn_wmma_f32_16x16x4_f32(false, A, false, B0, (short)0, z8, false, false);
      v8f W1 = __builtin_amdgcn_wmma_f32_16x16x4_f32(false, A, false, B1, (short)0, z8, false, false);
      v8f Zm = __builtin_amdgcn_wmma_f32_16x16x4_f32(false, A, false, B2, (short)0, z8, false, false);

      const int faceId = c * CHB + fl;
      #pragma unroll
      for (int q = 0; q < 8; ++q) {
        float w0 = W0[q], w1 = W1[q], z = Zm[q];
        float w2 = 1.0f - w0 - w1;
        float m  = fminf(fminf(w0, w1), w2);
        bool ok  = (m >= 0.0f) && (z > 0.0f);
        float zc = ok ? z : __builtin_inff();
        if (zc < zmin[q]) { zmin[q] = zc; fid[q] = faceId; }   // strict < keeps earliest face
      }
    }
    __syncthreads();                             // done reading buf[cur] before reuse
  }

  // per-pixel reduction across the 16 face columns (stays inside each half-wave)
  #pragma unroll
  for (int mask = 1; mask <= 8; mask <<= 1) {
    #pragma unroll
    for (int q = 0; q < 8; ++q) {
      float zo = __shfl_xor(zmin[q], mask, 32);
      int   io = __shfl_xor(fid[q],  mask, 32);
      if (zo < zmin[q] || (zo == zmin[q] && io < fid[q])) { zmin[q] = zo; fid[q] = io; }
    }
  }

  if (col == 0) {            // lane 0 -> pixels M=0..7, lane 16 -> pixels M=8..15
    int half = hi ? 8 : 0;
    #pragma unroll
    for (int q = 0; q < 8; ++q)
      fidOut[(size_t)b * P + basePix + half + q] = fid[q];
  }
}

// ---------------------------------------------------------------------------
// Kernel 3: shading. Recomputes barycentrics for the winning face from its
// coefficient record, clips, indexes the 4x4x4 texture, applies lighting.
// ---------------------------------------------------------------------------
__global__ __launch_bounds__(256) void nr_shade(
    const float* __restrict__ rec, const float* __restrict__ lightArr,
    const int* __restrict__ fidArr, const float* __restrict__ textures,
    float* __restrict__ out, int B, int Fpad)
{
  const int P = IMG * IMG;
  int i = blockIdx.x * blockDim.x + threadIdx.x;
  if (i >= B * P) return;
  int b = i / P, p = i - b * P;
  int fid = fidArr[i];
  float* img = out + (size_t)i * 3;
  float* sil = out + (size_t)B * P * 3 + i;
  if (fid < 0) { img[0] = 0.f; img[1] = 0.f; img[2] = 0.f; sil[0] = 0.f; return; }

  const float* r = rec + ((size_t)b * Fpad + fid) * 12;
  float px = (2.f * (float)(p & (IMG - 1)) + 1.f) * (1.f / IMG) - 1.f;
  float py = 1.f - (2.f * (float)(p >> 8) + 1.f) * (1.f / IMG);
  float w0 = r[0] * px + r[1] * py + r[6];
  float w1 = r[2] * px + r[3] * py + r[8];
  float w2 = 1.f - w0 - w1;
  w0 = fminf(fmaxf(w0, 0.f), 1.f);
  w1 = fminf(fmaxf(w1, 0.f), 1.f);
  w2 = fminf(fmaxf(w2, 0.f), 1.f);
  int t0 = (int)(w0 * 4.f); t0 = t0 < 0 ? 0 : (t0 > 3 ? 3 : t0);
  int t1 = (int)(w1 * 4.f); t1 = t1 < 0 ? 0 : (t1 > 3 ? 3 : t1);
  int t2 = (int)(w2 * 4.f); t2 = t2 < 0 ? 0 : (t2 > 3 ? 3 : t2);
  const float* tex = textures + ((size_t)fid * 64 + t0 * 16 + t1 * 4 + t2) * 3;
  float light = lightArr[(size_t)b * Fpad + fid];
  img[0] = tex[0] * light;
  img[1] = tex[1] * light;
  img[2] = tex[2] * light;
  sil[0] = 1.f;
}

// ---------------------------------------------------------------------------
extern "C" void kernel_launch(void* const* d_in, const int* in_sizes, int n_in,
                              void* d_out, int out_size, void* d_ws, size_t ws_size,
                              hipStream_t stream) {
  const float* verts    = (const float*)d_in[0];   // (B,V,3) f32
  const int*   faces    = (const int*)d_in[1];     // (F,3) i32
  const float* cams     = (const float*)d_in[2];   // (B,8) f32
  const float* textures = (const float*)d_in[3];   // (F,4,4,4,3) f32

  const int B = in_sizes[2] / 8;
  const int V = in_sizes[0] / (3 * B);
  const int F = in_sizes[1] / 3;
  const int Fpad = ((F + CHB - 1) / CHB) * CHB;    // all LDS chunks full
  const int P = IMG * IMG;

  // workspace layout
  float* rec      = (float*)d_ws;                         // B*Fpad*12 f32
  float* lightArr = rec + (size_t)B * Fpad * 12;          // B*Fpad f32
  int*   fidArr   = (int*)(lightArr + (size_t)B * Fpad);  // B*P i32

  int nPrep = B * Fpad;
  nr_prep<<<(nPrep + 255) / 256, 256, 0, stream>>>(verts, faces, cams, rec, lightArr,
                                                   B, V, F, Fpad);

  int groups = (B * P) / 16;        // one wave per 16-pixel group
  int blocks = groups / 8;          // 8 waves per 256-thread block
  nr_raster<<<blocks, 256, 0, stream>>>(rec, fidArr, Fpad);

  nr_shade<<<(B * P + 255) / 256, 256, 0, stream>>>(rec, lightArr, fidArr, textures,
                                                    (float*)d_out, B, Fpad);
}